// indi_SAGE_PR_1623497638161
// MI455X (gfx1250) — compile-verified
//
#include <hip/hip_runtime.h>

#define N_NODES 10000
#define N_EDGES 640000
#define IN_C    128
#define HID_C   256

typedef float v2f __attribute__((ext_vector_type(2)));
typedef float v8f __attribute__((ext_vector_type(8)));

// ---------------------------------------------------------------------------
// Degree count: one thread per edge, f32 atomic increment (L2-resident).
// ---------------------------------------------------------------------------
__global__ __launch_bounds__(256) void deg_kernel(const int* __restrict__ dst,
                                                  float* __restrict__ deg,
                                                  int n_edges) {
    int e = blockIdx.x * blockDim.x + threadIdx.x;
    if (e < n_edges) atomicAdd(&deg[dst[e]], 1.0f);
}

__global__ __launch_bounds__(256) void inv_deg_kernel(const float* __restrict__ deg,
                                                      float* __restrict__ inv,
                                                      int n) {
    int i = blockIdx.x * blockDim.x + threadIdx.x;
    if (i < n) inv[i] = 1.0f / fmaxf(deg[i], 1.0f);
}

// ---------------------------------------------------------------------------
// Edge scatter: one wave32 per edge; lanes stride the channel dimension.
// Gather source row (hits L2: feature matrix is only 5-10 MB) and atomically
// accumulate into agg[dst]. C is a compile-time constant (128 or 256).
// ---------------------------------------------------------------------------
template <int C>
__global__ __launch_bounds__(256) void scatter_add_kernel(
    const float* __restrict__ feat, const int* __restrict__ src,
    const int* __restrict__ dst, float* __restrict__ agg, int n_edges) {
    int t    = blockIdx.x * blockDim.x + threadIdx.x;
    int e    = t >> 5;
    int lane = t & 31;
    if (e >= n_edges) return;
    int s = src[e];
    int d = dst[e];
    const float* xs = feat + (size_t)s * C;
    float*       ad = agg  + (size_t)d * C;
#pragma unroll
    for (int c = lane; c < C; c += 32) {
        atomicAdd(ad + c, xs[c]);
    }
}

// ---------------------------------------------------------------------------
// Fused SAGE linear:  out = mean @ Wl + xin @ Wr + b  (optional ReLU)
// mean is formed on the fly as agg[row] * inv_deg[row] (folded into the A
// fragment). One wave computes a 16x16 output tile with chained
// v_wmma_f32_16x16x4_f32 accumulation (fp32 in / fp32 accumulate — matches
// the reference numerics; workload is memory-bound so low-precision WMMA
// buys nothing here).
//
// Fragment layouts per CDNA5 ISA 7.12.2 (wave32):
//   A (16x4, 2 VGPR):  lane l in half h holds A[M=l][K=2h+j] in VGPR j
//   B (4x16, 2 VGPR):  VGPR j, half h holds B[K=2h+j][N=l]
//   C/D (16x16, 8 VGPR): VGPR v, half h holds D[M=v+8h][N=l]
// ---------------------------------------------------------------------------
template <int CIN, bool RELU>
__global__ __launch_bounds__(256) void sage_gemm_kernel(
    const float* __restrict__ agg,      // [N, CIN]
    const float* __restrict__ inv_deg,  // [N]
    const float* __restrict__ xin,      // [N, CIN]
    const float* __restrict__ Wl,       // [CIN, COUT] row-major
    const float* __restrict__ Wr,       // [CIN, COUT] row-major
    const float* __restrict__ bias,     // [COUT]
    float* __restrict__ out,            // [N, COUT]
    int n_nodes, int cout) {
    const int wave   = (blockIdx.x * blockDim.x + threadIdx.x) >> 5;
    const int lane   = threadIdx.x & 31;
    const int ntiles = cout >> 4;
    const int mtile  = wave / ntiles;
    const int ntile  = wave - mtile * ntiles;
    const int row0   = mtile << 4;
    const int col0   = ntile << 4;
    if (row0 >= n_nodes) return;  // wave-uniform: EXEC stays all-1s below

    const int   h    = lane >> 4;   // lane half-group
    const int   l    = lane & 15;
    const int   arow = row0 + l;
    const float inv  = inv_deg[arow];

    const float* agg_row = agg + (size_t)arow * CIN;
    const float* x_row   = xin + (size_t)arow * CIN;

    v8f acc = {};
    for (int k = 0; k < CIN; k += 4) {
        const int ka = k + 2 * h;  // this lane-half's K pair base
        // A fragments (two consecutive K values per lane)
        v2f a_mean = *(const v2f*)(agg_row + ka);
        a_mean.x *= inv;
        a_mean.y *= inv;
        v2f a_x = *(const v2f*)(x_row + ka);
        // B fragments: column (col0+l) of rows ka, ka+1
        v2f b_l, b_r;
        b_l.x = Wl[(size_t)ka * cout + col0 + l];
        b_l.y = Wl[(size_t)(ka + 1) * cout + col0 + l];
        b_r.x = Wr[(size_t)ka * cout + col0 + l];
        b_r.y = Wr[(size_t)(ka + 1) * cout + col0 + l];
        // D = A*B + C chained accumulation on the matrix pipe
        acc = __builtin_amdgcn_wmma_f32_16x16x4_f32(
            false, a_mean, false, b_l, (short)0, acc, false, false);
        acc = __builtin_amdgcn_wmma_f32_16x16x4_f32(
            false, a_x, false, b_r, (short)0, acc, false, false);
    }

    const float bv = bias[col0 + l];
#pragma unroll
    for (int v = 0; v < 8; ++v) {
        int   r   = row0 + v + 8 * h;
        float val = acc[v] + bv;
        if (RELU) val = fmaxf(val, 0.0f);
        out[(size_t)r * cout + col0 + l] = val;
    }
}

// ---------------------------------------------------------------------------
// Launcher. Inputs (setup_inputs order):
//   0:x [N,128] f32   1:src [E] i32   2:dst [E] i32
//   3:W_l1 [128,256]  4:W_r1 [128,256]  5:b1 [256]
//   6:W_l2 [256,256]  7:W_r2 [256,256]  8:b2 [256]
// Output: [N,256] f32
// ---------------------------------------------------------------------------
extern "C" void kernel_launch(void* const* d_in, const int* in_sizes, int n_in,
                              void* d_out, int out_size, void* d_ws, size_t ws_size,
                              hipStream_t stream) {
    const float* x   = (const float*)d_in[0];
    const int*   src = (const int*)d_in[1];
    const int*   dst = (const int*)d_in[2];
    const float* Wl1 = (const float*)d_in[3];
    const float* Wr1 = (const float*)d_in[4];
    const float* b1  = (const float*)d_in[5];
    const float* Wl2 = (const float*)d_in[6];
    const float* Wr2 = (const float*)d_in[7];
    const float* b2  = (const float*)d_in[8];
    float*       out = (float*)d_out;

    // Workspace layout (bytes):
    //   agg1 [N*128 f32]  @ 0          (5,120,000)
    //   agg2 [N*256 f32]  @ 5,120,000  (10,240,000)
    //   h    [N*256 f32]  @ 15,360,000 (10,240,000)
    //   deg  [N f32]      @ 25,600,000 (40,000)
    //   inv  [N f32]      @ 25,640,000 (40,000)
    char*  ws   = (char*)d_ws;
    float* agg1 = (float*)(ws);
    float* agg2 = (float*)(ws + 5120000);
    float* hbuf = (float*)(ws + 15360000);
    float* deg  = (float*)(ws + 25600000);
    float* inv  = (float*)(ws + 25640000);

    // Zero accumulators every call (graph-capture safe async memset).
    hipMemsetAsync(agg1, 0, 15360000, stream);  // agg1 + agg2 (contiguous)
    hipMemsetAsync(deg, 0, 40000, stream);

    // Degrees + reciprocal (shared by both layers).
    deg_kernel<<<(N_EDGES + 255) / 256, 256, 0, stream>>>(dst, deg, N_EDGES);
    inv_deg_kernel<<<(N_NODES + 255) / 256, 256, 0, stream>>>(deg, inv, N_NODES);

    // Layer 1: scatter x -> agg1, then h = relu(mean1@Wl1 + x@Wr1 + b1)
    {
        const int threads = N_EDGES * 32;
        scatter_add_kernel<IN_C><<<threads / 256, 256, 0, stream>>>(
            x, src, dst, agg1, N_EDGES);
        const int waves  = (N_NODES / 16) * (HID_C / 16);  // 625*16 = 10000
        const int blocks = waves * 32 / 256;               // 1250
        sage_gemm_kernel<IN_C, true><<<blocks, 256, 0, stream>>>(
            agg1, inv, x, Wl1, Wr1, b1, hbuf, N_NODES, HID_C);
    }

    // Layer 2: scatter h -> agg2, then out = mean2@Wl2 + h@Wr2 + b2
    {
        const int threads = N_EDGES * 32;
        scatter_add_kernel<HID_C><<<threads / 256, 256, 0, stream>>>(
            hbuf, src, dst, agg2, N_EDGES);
        const int waves  = (N_NODES / 16) * (HID_C / 16);
        const int blocks = waves * 32 / 256;
        sage_gemm_kernel<HID_C, false><<<blocks, 256, 0, stream>>>(
            agg2, inv, hbuf, Wl2, Wr2, b2, out, N_NODES, HID_C);
    }
}